// EngineOrderFFT_36661840839137
// MI455X (gfx1250) — compile-verified
//
#include <hip/hip_runtime.h>
#include <math.h>

#define PI_F 3.14159265358979f

typedef __attribute__((ext_vector_type(2))) float v2f;
typedef __attribute__((ext_vector_type(8))) float v8f;

#define MFFT   16384   // Bluestein FFT length (2^14)
#define LSIG   8192    // input signal length / output bins (SF)
#define NCH    8
#define NBATCH 64
#define THREADS 512
#define NWAVES (THREADS / 32)

// ---------------------------------------------------------------------------
// 16-point DFT matrix fragments for V_WMMA_F32_16X16X4_F32.
// A-matrix (16x4 f32): lane l<16 holds {A[l][k0+0], A[l][k0+1]},
//                      lane l>=16 holds {A[l-16][k0+2], A[l-16][k0+3]}.
// We keep Fr, Fi, and -Fi (f32 WMMA has no A/B NEG modifier).
// ---------------------------------------------------------------------------
struct F16Frag {
  v2f fr[4];
  v2f fi[4];
  v2f fin[4];
};

__device__ __forceinline__ void make_dft16_frags(F16Frag& f) {
  const int lane = threadIdx.x & 31;
  const int m = lane & 15;
  const int h = lane >> 4;
#pragma unroll
  for (int c = 0; c < 4; ++c) {
#pragma unroll
    for (int j = 0; j < 2; ++j) {
      const int k = 4 * c + 2 * h + j;
      const float ang = -2.0f * PI_F * (float)((m * k) & 15) / 16.0f;
      float sn, cs;
      __sincosf(ang, &sn, &cs);
      f.fr[c][j]  = cs;   // Re W16^{mk}
      f.fi[c][j]  = sn;   // Im W16^{mk}  (W16 = e^{-2*pi*i/16})
      f.fin[c][j] = -sn;
    }
  }
}

// Complex 16x16 = DFT16 x (16 columns), via 16 f32 WMMAs.
__device__ __forceinline__ void cmatmul16(const F16Frag& f,
                                          const v2f br[4], const v2f bi[4],
                                          v8f& yr, v8f& yi) {
  v8f ar = {0.f, 0.f, 0.f, 0.f, 0.f, 0.f, 0.f, 0.f};
  v8f ai = {0.f, 0.f, 0.f, 0.f, 0.f, 0.f, 0.f, 0.f};
#pragma unroll
  for (int c = 0; c < 4; ++c)
    ar = __builtin_amdgcn_wmma_f32_16x16x4_f32(false, f.fr[c],  false, br[c],
                                               (short)0, ar, false, false);
#pragma unroll
  for (int c = 0; c < 4; ++c)
    ar = __builtin_amdgcn_wmma_f32_16x16x4_f32(false, f.fin[c], false, bi[c],
                                               (short)0, ar, false, false);
#pragma unroll
  for (int c = 0; c < 4; ++c)
    ai = __builtin_amdgcn_wmma_f32_16x16x4_f32(false, f.fr[c],  false, bi[c],
                                               (short)0, ai, false, false);
#pragma unroll
  for (int c = 0; c < 4; ++c)
    ai = __builtin_amdgcn_wmma_f32_16x16x4_f32(false, f.fi[c],  false, br[c],
                                               (short)0, ai, false, false);
  yr = ar;
  yi = ai;
}

// ---------------------------------------------------------------------------
// Radix-16 stage (stride s = 1<<sh, block size S = 16s), in-place in LDS.
// forward (DIF):  butterfly, then twiddle W_S^{q*r} on outputs.
// inverse (transposed stage): twiddle W_S^{p*r} on inputs, then butterfly.
// Each wave processes 16 DFTs (16 matrix columns) per WMMA macro-op.
// ---------------------------------------------------------------------------
__device__ void stage16(float* lre, float* lim, int sh, bool inverse_side,
                        const F16Frag& f) {
  const int s    = 1 << sh;
  const int Ssz  = s << 4;
  const int wave = threadIdx.x >> 5;
  const int lane = threadIdx.x & 31;
  const int col  = lane & 15;
  const int h    = lane >> 4;

#pragma unroll 1
  for (int it = wave; it < (MFFT / 16) / 16; it += NWAVES) {  // 64 macro-ops
    const int g    = it * 16 + col;       // DFT id handled by this column
    const int bl   = g >> sh;
    const int r    = g & (s - 1);
    const int base = (bl << (sh + 4)) + r;

    v2f br[4], bi[4];
#pragma unroll
    for (int c = 0; c < 4; ++c) {
#pragma unroll
      for (int j = 0; j < 2; ++j) {
        const int p = 4 * c + 2 * h + j;
        const int addr = base + (p << sh);
        float xr = lre[addr];
        float xi = lim[addr];
        if (inverse_side) {
          const float ang =
              -2.0f * PI_F * (float)((p * r) & (Ssz - 1)) / (float)Ssz;
          float sn, cs;
          __sincosf(ang, &sn, &cs);
          const float tr = xr * cs - xi * sn;
          const float ti = xr * sn + xi * cs;
          xr = tr; xi = ti;
        }
        br[c][j] = xr;
        bi[c][j] = xi;
      }
    }

    v8f yr, yi;
    cmatmul16(f, br, bi, yr, yi);

#pragma unroll
    for (int vr = 0; vr < 8; ++vr) {
      const int q = vr + 8 * h;
      float wr = yr[vr], wi = yi[vr];
      if (!inverse_side) {
        const float ang =
            -2.0f * PI_F * (float)((q * r) & (Ssz - 1)) / (float)Ssz;
        float sn, cs;
        __sincosf(ang, &sn, &cs);
        const float tr = wr * cs - wi * sn;
        const float ti = wr * sn + wi * cs;
        wr = tr; wi = ti;
      }
      const int addr = base + (q << sh);
      lre[addr] = wr;
      lim[addr] = wi;
    }
  }
}

// Final/first radix-4 stage, stride 1, no twiddles (W_4^{q*0}=1). Symmetric,
// so identical for forward and transposed-inverse.
__device__ void stage4(float* lre, float* lim) {
#pragma unroll 1
  for (int gidx = threadIdx.x; gidx < MFFT / 4; gidx += THREADS) {
    const int base = gidx * 4;
    const float x0r = lre[base + 0], x0i = lim[base + 0];
    const float x1r = lre[base + 1], x1i = lim[base + 1];
    const float x2r = lre[base + 2], x2i = lim[base + 2];
    const float x3r = lre[base + 3], x3i = lim[base + 3];
    const float e0r = x0r + x2r, e0i = x0i + x2i;
    const float e1r = x0r - x2r, e1i = x0i - x2i;
    const float o0r = x1r + x3r, o0i = x1i + x3i;
    const float o1r = x1r - x3r, o1i = x1i - x3i;
    lre[base + 0] = e0r + o0r; lim[base + 0] = e0i + o0i;   // y0 = e0 + o0
    lre[base + 2] = e0r - o0r; lim[base + 2] = e0i - o0i;   // y2 = e0 - o0
    lre[base + 1] = e1r + o1i; lim[base + 1] = e1i - o1r;   // y1 = e1 - i*o1
    lre[base + 3] = e1r - o1i; lim[base + 3] = e1i + o1r;   // y3 = e1 + i*o1
  }
}

__device__ void fft_forward(float* lre, float* lim, const F16Frag& f) {
  stage16(lre, lim, 10, false, f); __syncthreads();
  stage16(lre, lim, 6,  false, f); __syncthreads();
  stage16(lre, lim, 2,  false, f); __syncthreads();
  stage4(lre, lim);                __syncthreads();
}

// Consumes the digit-scrambled order produced by fft_forward, emits natural.
__device__ void fft_transposed(float* lre, float* lim, const F16Frag& f) {
  stage4(lre, lim);                __syncthreads();
  stage16(lre, lim, 2,  true, f);  __syncthreads();
  stage16(lre, lim, 6,  true, f);  __syncthreads();
  stage16(lre, lim, 10, true, f);  __syncthreads();
}

// ---------------------------------------------------------------------------
// Kernel 1: per-batch scrambled fft(b) of the Bluestein chirp into workspace.
// ws layout: [0, 64*16384)        -> Bhat real
//            [64*16384, 2*64*16384) -> Bhat imag
// ---------------------------------------------------------------------------
__global__ void __launch_bounds__(THREADS)
bchirp_fft_kernel(const float* __restrict__ rpm, float* __restrict__ ws) {
  __shared__ float lre[MFFT];
  __shared__ float lim[MFFT];
  const int b = blockIdx.x;
  const float rv = rpm[b];
  const int pad = (int)floorf((2400.0f / rv - 1.0f) * 8192.0f);
  const int n = LSIG + pad;
  const int two_n = 2 * n;
  const float nf = (float)n;

  for (int m = threadIdx.x; m < MFFT; m += THREADS) {
    const int mm = (m < MFFT - m) ? m : (MFFT - m);
    const int t2 = (mm * mm) % two_n;
    const float ang = PI_F * (float)t2 / nf;  // b = exp(+i*pi*mm^2/n)
    float sn, cs;
    __sincosf(ang, &sn, &cs);
    lre[m] = cs;
    lim[m] = sn;
  }
  __syncthreads();

  F16Frag f;
  make_dft16_frags(f);
  fft_forward(lre, lim, f);

  float* bre = ws + (size_t)b * MFFT;
  float* bim = ws + (size_t)NBATCH * MFFT + (size_t)b * MFFT;
  for (int m = threadIdx.x; m < MFFT; m += THREADS) {
    bre[m] = lre[m];
    bim[m] = lim[m];
  }
}

// ---------------------------------------------------------------------------
// Kernel 2: one (batch, channel) per block.
//   a = x * exp(-i*pi*t^2/n)  -> forward FFT (scrambled)
//   u = conj(A_hat * B_hat)   (pointwise, scrambled order)
//   w = F(u)  via transposed stages (natural order) ; out = |w| / M
//   (final chirp has |.|==1, so it drops out under abs)
// ---------------------------------------------------------------------------
__global__ void __launch_bounds__(THREADS)
eofft_main_kernel(const float* __restrict__ x, const float* __restrict__ rpm,
                  const float* __restrict__ ws, float* __restrict__ out) {
  __shared__ float lre[MFFT];
  __shared__ float lim[MFFT];
  const int b = blockIdx.x >> 3;
  const int c = blockIdx.x & 7;
  const float rv = rpm[b];
  const int pad = (int)floorf((2400.0f / rv - 1.0f) * 8192.0f);
  const int n = LSIG + pad;
  const int two_n = 2 * n;
  const float nf = (float)n;

  const float* xb = x + (size_t)b * LSIG * NCH + c;
  for (int t = threadIdx.x; t < MFFT; t += THREADS) {
    if (t < LSIG) {
      const float xv = xb[(size_t)t * NCH];
      const int t2 = (t * t) % two_n;
      const float ang = -PI_F * (float)t2 / nf;
      float sn, cs;
      __sincosf(ang, &sn, &cs);
      lre[t] = xv * cs;
      lim[t] = xv * sn;
    } else {
      lre[t] = 0.0f;
      lim[t] = 0.0f;
    }
  }
  __syncthreads();

  F16Frag f;
  make_dft16_frags(f);
  fft_forward(lre, lim, f);

  const float* bre = ws + (size_t)b * MFFT;
  const float* bim = ws + (size_t)NBATCH * MFFT + (size_t)b * MFFT;
  for (int m = threadIdx.x; m < MFFT; m += THREADS) {
    const float ar = lre[m], ai = lim[m];
    const float brv = bre[m], biv = bim[m];
    lre[m] = ar * brv - ai * biv;       // Re(A*B)
    lim[m] = -(ar * biv + ai * brv);    // conj
  }
  __syncthreads();

  fft_transposed(lre, lim, f);

  const float scale = 1.0f / (float)MFFT;
  float* ob = out + (size_t)b * LSIG * NCH + c;
  for (int k = threadIdx.x; k < LSIG; k += THREADS) {
    const float wr = lre[k], wi = lim[k];
    ob[(size_t)k * NCH] = scale * sqrtf(wr * wr + wi * wi);
  }
}

extern "C" void kernel_launch(void* const* d_in, const int* in_sizes, int n_in,
                              void* d_out, int out_size, void* d_ws,
                              size_t ws_size, hipStream_t stream) {
  const float* x   = (const float*)d_in[0];  // (64, 8192, 8) f32
  const float* rpm = (const float*)d_in[1];  // (64,) f32
  float* ws  = (float*)d_ws;                 // needs 2*64*16384*4 = 8 MB
  float* out = (float*)d_out;                // (64, 8192, 8) f32

  bchirp_fft_kernel<<<NBATCH, THREADS, 0, stream>>>(rpm, ws);
  eofft_main_kernel<<<NBATCH * NCH, THREADS, 0, stream>>>(x, rpm, ws, out);
}